// PostProcessSMCA_57200374448167
// MI455X (gfx1250) — compile-verified
//
#include <hip/hip_runtime.h>
#include <hip/hip_bf16.h>
#include <stdint.h>

// Problem dims
#define BS 256
#define NQ 1000
#define NC 92
#define NEL (NQ * NC)                 // 92000 logits per batch row
#define NTHREADS 1024                 // 32 wave32s per workgroup
#define TILE_FLOATS (NTHREADS * 4)    // 4096 floats = 16 KB per streamed chunk
#define NCHUNK ((NEL + TILE_FLOATS - 1) / TILE_FLOATS)   // 23
#define SORTN 1024                    // bitonic sort size (k=1000 padded)

typedef int v4i __attribute__((ext_vector_type(4)));
typedef __attribute__((address_space(1))) v4i gv4i_t;   // global int4*
typedef __attribute__((address_space(3))) v4i lv4i_t;   // LDS int4*
typedef __attribute__((address_space(3))) void lvoid_t;

// ---- CDNA5 async global -> LDS DMA (ASYNCcnt-tracked) -------------------
__device__ __forceinline__ void async_copy16(const float* g, float* l) {
#if __has_builtin(__builtin_amdgcn_global_load_async_to_lds_b128)
  __builtin_amdgcn_global_load_async_to_lds_b128((gv4i_t*)g, (lv4i_t*)l, 0, 0);
#else
  unsigned loff = (unsigned)(uintptr_t)(lvoid_t*)l;         // 32-bit LDS addr
  unsigned long long ga = (unsigned long long)(uintptr_t)g; // 64-bit global addr
  asm volatile("global_load_async_to_lds_b128 %0, %1, off"
               :: "v"(loff), "v"(ga) : "memory");
#endif
}

__device__ __forceinline__ void wait_async0() {
#if __has_builtin(__builtin_amdgcn_s_wait_asynccnt)
  __builtin_amdgcn_s_wait_asynccnt(0);
#else
  asm volatile("s_wait_asynccnt 0" ::: "memory");
#endif
}

__device__ __forceinline__ void wait_async1() {
#if __has_builtin(__builtin_amdgcn_s_wait_asynccnt)
  __builtin_amdgcn_s_wait_asynccnt(1);
#else
  asm volatile("s_wait_asynccnt 1" ::: "memory");
#endif
}

// Monotone float-bits <-> uint mapping (order-preserving, so top-k on raw
// logits == top-k on sigmoid(logits); sigmoid computed only for winners).
__device__ __forceinline__ unsigned monot(unsigned u) {
  return (u & 0x80000000u) ? ~u : (u | 0x80000000u);
}
__device__ __forceinline__ unsigned unmonot(unsigned k) {
  return (k & 0x80000000u) ? (k ^ 0x80000000u) : ~k;
}

__global__ __launch_bounds__(NTHREADS)
void PostProcessSMCA_topk_kernel(const float* __restrict__ logits,
                                 const float* __restrict__ boxes,
                                 const float* __restrict__ sizes,
                                 float* __restrict__ out_scores,
                                 float* __restrict__ out_labels,
                                 float* __restrict__ out_boxes) {
  __shared__ float stage[2][TILE_FLOATS];          // 32 KB double-buffer
  __shared__ unsigned hist[256];                   // 1 KB
  __shared__ unsigned long long keys[SORTN];       // 8 KB packed (key,~idx)
  __shared__ unsigned s_tbin, s_above, s_ctrA, s_ctrB;

  const int b = blockIdx.x;
  const unsigned t = threadIdx.x;
  const float* row = logits + (size_t)b * NEL;

  auto issue = [&](int ch, int buf) {
    int idx4 = ch * TILE_FLOATS + (int)t * 4;
    if (idx4 < NEL) async_copy16(row + idx4, &stage[buf][t * 4]);
  };

  // ---------------- Pass 1: streamed 256-bin histogram -------------------
  if (t < 256) hist[t] = 0u;
  __syncthreads();

  issue(0, 0);
  for (int ch = 0; ch < NCHUNK; ++ch) {
    const int cur = ch & 1;
    if (ch + 1 < NCHUNK) { issue(ch + 1, cur ^ 1); wait_async1(); }
    else                 { wait_async0(); }
    __syncthreads();
    const int idx4 = ch * TILE_FLOATS + (int)t * 4;
    if (idx4 < NEL) {
      float4 v = *(const float4*)&stage[cur][t * 4];
      atomicAdd(&hist[monot(__float_as_uint(v.x)) >> 24], 1u);
      atomicAdd(&hist[monot(__float_as_uint(v.y)) >> 24], 1u);
      atomicAdd(&hist[monot(__float_as_uint(v.z)) >> 24], 1u);
      atomicAdd(&hist[monot(__float_as_uint(v.w)) >> 24], 1u);
    }
    __syncthreads();
  }

  // ---------------- Threshold bin (scan from top) ------------------------
  if (t == 0) {
    unsigned cum = 0;
    int bin = 255;
    for (; bin > 0; --bin) {
      unsigned c = hist[bin];
      if (cum + c >= NQ) break;
      cum += c;
    }
    s_tbin = (unsigned)bin;
    s_above = cum;     // count strictly above threshold bin (< NQ)
    s_ctrA = 0u;
    s_ctrB = 0u;
  }
  for (unsigned i = t; i < SORTN; i += NTHREADS) keys[i] = 0ull;  // pad = -inf
  __syncthreads();
  const unsigned tbin = s_tbin;
  const unsigned above = s_above;

  // ---------------- Pass 2: streamed candidate collection ----------------
  issue(0, 0);
  for (int ch = 0; ch < NCHUNK; ++ch) {
    const int cur = ch & 1;
    if (ch + 1 < NCHUNK) { issue(ch + 1, cur ^ 1); wait_async1(); }
    else                 { wait_async0(); }
    __syncthreads();
    const int idx4 = ch * TILE_FLOATS + (int)t * 4;
    if (idx4 < NEL) {
      float4 v = *(const float4*)&stage[cur][t * 4];
      float a[4] = {v.x, v.y, v.z, v.w};
#pragma unroll
      for (int j = 0; j < 4; ++j) {
        unsigned key = monot(__float_as_uint(a[j]));
        unsigned bin = key >> 24;
        if (bin >= tbin) {
          unsigned slot = (bin > tbin) ? atomicAdd(&s_ctrA, 1u)
                                       : above + atomicAdd(&s_ctrB, 1u);
          if (slot < NQ) {
            unsigned gidx = (unsigned)(idx4 + j);
            keys[slot] = ((unsigned long long)key << 32) |
                         (unsigned long long)(~gidx);   // ~idx: small idx first
          }
        }
      }
    }
    __syncthreads();
  }

  // ---------------- Bitonic sort, descending, 1024 elems in LDS ----------
  for (unsigned sz = 2; sz <= SORTN; sz <<= 1) {
    for (unsigned stride = sz >> 1; stride > 0; stride >>= 1) {
      __syncthreads();
      unsigned i = t;
      unsigned j = i ^ stride;
      if (j > i) {
        unsigned long long x = keys[i], y = keys[j];
        bool desc = ((i & sz) == 0);
        if (desc ? (x < y) : (x > y)) { keys[i] = y; keys[j] = x; }
      }
    }
  }
  __syncthreads();

  // ---------------- Emit: scores, labels, gathered+scaled boxes ----------
  if (t < NQ) {
    unsigned long long kv = keys[t];
    unsigned key  = (unsigned)(kv >> 32);
    unsigned gidx = ~((unsigned)(kv & 0xFFFFFFFFull));
    float logit = __uint_as_float(unmonot(key));
    float score = 1.0f / (1.0f + __expf(-logit));
    unsigned qi = gidx / NC;
    unsigned label = gidx - qi * NC;

    const size_t o = (size_t)b * NQ + t;
    out_scores[o] = score;
    out_labels[o] = (float)label;

    const float* bx = boxes + ((size_t)b * NQ + qi) * 4;
    float cx = bx[0], cy = bx[1], w = bx[2], h = bx[3];
    float img_h = sizes[2 * b + 0], img_w = sizes[2 * b + 1];
    float* ob = out_boxes + o * 4;
    ob[0] = (cx - 0.5f * w) * img_w;
    ob[1] = (cy - 0.5f * h) * img_h;
    ob[2] = (cx + 0.5f * w) * img_w;
    ob[3] = (cy + 0.5f * h) * img_h;
  }
}

extern "C" void kernel_launch(void* const* d_in, const int* in_sizes, int n_in,
                              void* d_out, int out_size, void* d_ws, size_t ws_size,
                              hipStream_t stream) {
  const float* logits = (const float*)d_in[0];   // (256,1000,92) f32
  const float* boxes  = (const float*)d_in[1];   // (256,1000,4)  f32
  const float* sizes  = (const float*)d_in[2];   // (256,2)       f32
  float* out = (float*)d_out;                    // scores ++ labels ++ boxes
  float* out_scores = out;
  float* out_labels = out + (size_t)BS * NQ;
  float* out_boxes  = out + (size_t)BS * NQ * 2;

  PostProcessSMCA_topk_kernel<<<BS, NTHREADS, 0, stream>>>(
      logits, boxes, sizes, out_scores, out_labels, out_boxes);
}